// GatedRepulsionPotential_81509889343867
// MI455X (gfx1250) — compile-verified
//
#include <hip/hip_runtime.h>

typedef __attribute__((ext_vector_type(2))) float v2f;
typedef __attribute__((ext_vector_type(8))) float v8f;

#define DIM 64
#define EPS 0.1f

__device__ __forceinline__ void atomic_add_f32(float* p, float v) {
  unsafeAtomicAdd(p, v);  // hardware global_atomic_add_f32 (L2 RMW)
}

__global__ void __launch_bounds__(256) zero_f32_kernel(float* __restrict__ p, size_t n) {
  size_t i = (size_t)blockIdx.x * blockDim.x + threadIdx.x;
  if (i < n) p[i] = 0.0f;
}

// One wave32 handles a tile of 16 edges.
//   A fragment (16x4 f32, per K-step): lane l<16 holds XU[l][kb+0..1], lane l>=16 holds XU[l-16][kb+2..3]
//   B fragment (4x16 f32): symmetric transpose layout, holds (J*XV)^T
// Gathers land directly in fragment layout -> no cross-lane shuffles needed.
__global__ void __launch_bounds__(256) hyper_repulse_kernel(
    const float* __restrict__ x, const int* __restrict__ u_idx,
    const int* __restrict__ v_idx, float* __restrict__ out, size_t E) {
  const int lane = threadIdx.x & 31;
  const int wave = threadIdx.x >> 5;
  const size_t e0 = ((size_t)blockIdx.x * 8 + wave) * 16;
  if (e0 >= E) return;  // uniform per wave -> WMMA EXEC stays all-ones

  float* __restrict__ grad = out + 1;

  const int m   = lane & 15;            // edge-in-tile owned (half-row) by this lane
  const int off = (lane < 16) ? 0 : 2;  // K sub-column pair
  size_t eg = e0 + (size_t)m;
  const bool valid = (eg < E);
  if (!valid) eg = E - 1;               // clamped gather; contribution zeroed below

  const int ui = u_idx[eg];
  const int vi = v_idx[eg];
  const float* __restrict__ xu = x + (size_t)ui * DIM;
  const float* __restrict__ xv = x + (size_t)vi * DIM;

  v2f afrag[16], bfrag[16];
#pragma unroll
  for (int kk = 0; kk < 16; ++kk) {
    v2f a = *(const v2f*)(xu + kk * 4 + off);
    v2f b = *(const v2f*)(xv + kk * 4 + off);
    // Minkowski J on column 0 (B carries xv*J); branchless, EXEC untouched
    b.x = (kk == 0 && lane < 16) ? -b.x : b.x;
    afrag[kk] = a;
    bfrag[kk] = b;
  }

  // D(16x16) = XU(16x64) x (J*XV)^T(64x16), 16 steps of K=4. Diagonal = inner[e].
  v8f c = {};
#pragma unroll
  for (int kk = 0; kk < 16; ++kk) {
    c = __builtin_amdgcn_wmma_f32_16x16x4_f32(false, afrag[kk],
                                              false, bfrag[kk],
                                              (short)0, c, false, false);
  }

  // Diagonal D[e][e]: e<8 -> lane e, vgpr e ; e>=8 -> lane e+16, vgpr e-8.
  const int r = lane & 7;
  float t01 = (r & 1) ? c[1] : c[0];
  float t23 = (r & 1) ? c[3] : c[2];
  float t45 = (r & 1) ? c[5] : c[4];
  float t67 = (r & 1) ? c[7] : c[6];
  float t03 = (r & 2) ? t23 : t01;
  float t47 = (r & 2) ? t67 : t45;
  float inner = (r & 4) ? t47 : t03;   // meaningful on owner lanes {0..7, 24..31}

  const bool owner = (lane < 8) || (lane >= 24);

  float ic    = fminf(inner, -1.0f - 1e-7f);
  float z     = -ic;
  float denom = sqrtf(z * z - 1.0f);   // = sqrt(inner^2 - 1), same cancellation as ref
  float dist  = logf(z + denom);       // arccosh(z)
  bool  act   = (dist < EPS) && valid && owner;
  float delta = act ? (EPS - dist) : 0.0f;
  float factor_o = act ? (-(10.0f * delta) / (denom + 1e-9f)) : 0.0f;  // A/NUM_NEG = 10
  float eloc  = 5.0f * delta * delta;  // 0.5 * A / NUM_NEG = 5

  // Broadcast factor for edge m from its owner lane (ds_bpermute under the hood).
  const int src = (m < 8) ? m : (m + 16);
  float factor = __shfl(factor_o, src, 32);

  if (factor != 0.0f) {  // skip inactive edges' atomics (~45% of traffic)
    float* gu = grad + (size_t)ui * DIM;
    float* gv = grad + (size_t)vi * DIM;
#pragma unroll
    for (int kk = 0; kk < 16; ++kk) {
      const int col = kk * 4 + off;
      v2f a = afrag[kk];
      v2f b = bfrag[kk];
      float axJ = (kk == 0 && lane < 16) ? -a.x : a.x;   // xu*J
      atomic_add_f32(&gu[col],     factor * b.x);        // b already xv*J
      atomic_add_f32(&gu[col + 1], factor * b.y);
      atomic_add_f32(&gv[col],     factor * axJ);
      atomic_add_f32(&gv[col + 1], factor * a.y);
    }
  }

  // Energy: one atomic per wave
#pragma unroll
  for (int s = 16; s > 0; s >>= 1) eloc += __shfl_xor(eloc, s, 32);
  if (lane == 0 && eloc != 0.0f) atomic_add_f32(&out[0], eloc);
}

extern "C" void kernel_launch(void* const* d_in, const int* in_sizes, int n_in,
                              void* d_out, int out_size, void* d_ws, size_t ws_size,
                              hipStream_t stream) {
  (void)n_in; (void)d_ws; (void)ws_size;
  const float* x     = (const float*)d_in[0];
  const int*   u_idx = (const int*)d_in[1];
  const int*   v_idx = (const int*)d_in[2];
  float*       out   = (float*)d_out;

  const size_t E   = (size_t)in_sizes[1];        // N * NUM_NEG edges
  const size_t tot = (size_t)out_size;           // 1 + N*DIM

  // Harness poisons d_out: zero energy + grad first (same stream => ordered).
  const int zb = (int)((tot + 255) / 256);
  zero_f32_kernel<<<zb, 256, 0, stream>>>(out, tot);

  // 8 waves/block * 16 edges/wave = 128 edges per block.
  const int blocks = (int)((E + 127) / 128);
  hyper_repulse_kernel<<<blocks, 256, 0, stream>>>(x, u_idx, v_idx, out, E);
}